// GATEncoder_26723286515823
// MI455X (gfx1250) — compile-verified
//
#include <hip/hip_runtime.h>
#include <hip/hip_bf16.h>
#include <math.h>

typedef float v2f __attribute__((ext_vector_type(2)));
typedef float v8f __attribute__((ext_vector_type(8)));

#define NEG_SLOPE 0.2f
#define D1 128            // HEADS*HID
#define HID_C 64

// ---------- helpers ----------
__device__ __forceinline__ unsigned f2mono(float f) {
    unsigned u = __float_as_uint(f);
    return (u & 0x80000000u) ? ~u : (u | 0x80000000u);
}
__device__ __forceinline__ float mono2f(unsigned k) {
    unsigned u = (k & 0x80000000u) ? (k & 0x7FFFFFFFu) : ~k;
    return __uint_as_float(u);
}
__device__ __forceinline__ void edge_nodes(const int* __restrict__ ei, long long e,
                                           long long E, int& s, int& d) {
    if (e < E) { s = ei[e]; d = ei[E + e]; }
    else       { s = d = (int)(e - E); }        // self-loop
}
__device__ __forceinline__ float lrelu(float v) {
    return v > 0.f ? v : NEG_SLOPE * v;
}

// CDNA5 async global->LDS copy (ASYNCcnt-tracked, no VGPR round trip).
__device__ __forceinline__ void async_copy_b128(unsigned lds_byte_off,
                                                const void* gaddr) {
    asm volatile("global_load_async_to_lds_b128 %0, %1, off"
                 :: "v"(lds_byte_off),
                    "v"((unsigned long long)(uintptr_t)gaddr)
                 : "memory");
}
__device__ __forceinline__ void async_wait0() {
    asm volatile("s_wait_asynccnt 0x0" ::: "memory");
}

// ---------- 1) H = X(N x 128) * W(128 x 128), fp32 WMMA, async-LDS staged ----------
__global__ __launch_bounds__(256)
void gemm128_wmma(const float* __restrict__ X, const float* __restrict__ W,
                  float* __restrict__ H, int N)
{
    __shared__ float Wl[64 * 128];   // 32 KB: W rows [p*64, p*64+64), row-major 64x128
    __shared__ float Xl[128 * 64];   // 32 KB: X rows [blk*128, +128) cols [p*64, +64)

    const int tid  = threadIdx.x;
    const int wave = tid >> 5;
    const int lane = tid & 31;
    const int m    = lane & 15;
    const int hi   = lane >> 4;                    // 0 or 1
    const int kof  = hi * 2;                       // fp32 frag K sub-offset
    const long long blkRow0 = (long long)blockIdx.x * 128;
    const long long rowBase = blkRow0 + wave * 16;

    const unsigned WlBase = (unsigned)(uintptr_t)(void*)Wl;
    const unsigned XlBase = (unsigned)(uintptr_t)(void*)Xl;

    v8f acc[8] = {};                               // 8 column tiles of 16

    for (int p = 0; p < 2; ++p) {
        __syncthreads();                           // prior phase done reading LDS

        // --- stage W rows [p*64, p*64+64): 2048 float4, linear, async ---
        const char* wg = (const char*)(W + (size_t)p * 64 * 128);
        #pragma unroll
        for (int i = 0; i < 8; ++i) {
            unsigned boff = (unsigned)(i * 256 + tid) * 16u;
            async_copy_b128(WlBase + boff, wg + boff);
        }
        // --- stage X block rows [blkRow0, +128) cols [p*64, +64): 2048 float4 ---
        #pragma unroll
        for (int i = 0; i < 8; ++i) {
            int f = i * 256 + tid;                 // float4 index: row = f>>4, col4 = f&15
            long long r = blkRow0 + (f >> 4);
            if (r >= N) r = (long long)N - 1;      // clamp in-bounds; OOB rows never stored
            const float* gp = X + r * 128 + (size_t)p * 64 + (f & 15) * 4;
            async_copy_b128(XlBase + (unsigned)f * 16u, gp);
        }
        async_wait0();                             // own copies landed in LDS
        __syncthreads();                           // everyone's copies visible

        const int xrow = wave * 16 + m;            // A-frag row within block
        #pragma unroll
        for (int kk = 0; kk < 64; kk += 4) {
            v2f a;
            const float2 av = *(const float2*)&Xl[xrow * 64 + kk + kof];  // one ds_load_b64
            a.x = av.x; a.y = av.y;
            #pragma unroll
            for (int t = 0; t < 8; ++t) {
                v2f b;
                b.x = Wl[(kk + kof    ) * 128 + t * 16 + m];
                b.y = Wl[(kk + kof + 1) * 128 + t * 16 + m];
                acc[t] = __builtin_amdgcn_wmma_f32_16x16x4_f32(
                            false, a, false, b, (short)0, acc[t], false, false);
            }
        }
    }

    #pragma unroll
    for (int g = 0; g < 8; ++g) {
        long long r = rowBase + g + hi * 8;
        if (r < (long long)N) {
            #pragma unroll
            for (int t = 0; t < 8; ++t)
                H[r * 128 + t * 16 + m] = acc[t][g];
        }
    }
}

// ---------- 2) per-node attention logits ----------
__global__ __launch_bounds__(256)
void node_alpha(const float* __restrict__ H, const float* __restrict__ Asrc,
                const float* __restrict__ Adst, float* __restrict__ as_,
                float* __restrict__ ad_, int N)
{
    int n = blockIdx.x * blockDim.x + threadIdx.x;
    if (n >= N) return;
    const float4* h4 = (const float4*)(H + (long long)n * 128);
    const float4* s4 = (const float4*)Asrc;
    const float4* d4 = (const float4*)Adst;
    #pragma unroll
    for (int head = 0; head < 2; ++head) {
        float ss = 0.f, dd = 0.f;
        #pragma unroll
        for (int i = 0; i < 16; ++i) {
            float4 v = h4[head * 16 + i];
            float4 a = s4[head * 16 + i];
            float4 b = d4[head * 16 + i];
            ss += v.x * a.x + v.y * a.y + v.z * a.z + v.w * a.w;
            dd += v.x * b.x + v.y * b.y + v.z * b.z + v.w * b.w;
        }
        as_[(long long)n * 2 + head] = ss;
        ad_[(long long)n * 2 + head] = dd;
    }
}

// ---------- 3) segment max of leaky-relu logits (monotone-uint atomic max) ----------
__global__ __launch_bounds__(256)
void edge_max(const int* __restrict__ ei, const float* __restrict__ as_,
              const float* __restrict__ ad_, unsigned* __restrict__ mk,
              long long E, long long Etot)
{
    long long e = (long long)blockIdx.x * blockDim.x + threadIdx.x;
    if (e >= Etot) return;
    int s, d; edge_nodes(ei, e, E, s, d);
    #pragma unroll
    for (int h = 0; h < 2; ++h) {
        float v = lrelu(as_[(long long)s * 2 + h] + ad_[(long long)d * 2 + h]);
        atomicMax(&mk[(long long)d * 2 + h], f2mono(v));
    }
}

// ---------- 4) segment sum of exp(e - m) ----------
__global__ __launch_bounds__(256)
void edge_sum(const int* __restrict__ ei, const float* __restrict__ as_,
              const float* __restrict__ ad_, const unsigned* __restrict__ mk,
              float* __restrict__ ssum, long long E, long long Etot)
{
    long long e = (long long)blockIdx.x * blockDim.x + threadIdx.x;
    if (e >= Etot) return;
    int s, d; edge_nodes(ei, e, E, s, d);
    #pragma unroll
    for (int h = 0; h < 2; ++h) {
        float v = lrelu(as_[(long long)s * 2 + h] + ad_[(long long)d * 2 + h]);
        float m = mono2f(mk[(long long)d * 2 + h]);
        unsafeAtomicAdd(&ssum[(long long)d * 2 + h], __expf(v - m));
    }
}

// ---------- 5) message aggregation: one wave32 per edge ----------
__global__ __launch_bounds__(256)
void edge_agg(const int* __restrict__ ei, const float* __restrict__ as_,
              const float* __restrict__ ad_, const unsigned* __restrict__ mk,
              const float* __restrict__ ssum, const float* __restrict__ H,
              float* __restrict__ out, long long E, long long Etot)
{
    const int lane = threadIdx.x & 31;
    long long e = (long long)blockIdx.x * 8 + (threadIdx.x >> 5);
    if (e >= Etot) return;
    int s, d; edge_nodes(ei, e, E, s, d);
    const int head = lane >> 4;                    // channels lane*4..lane*4+3
    float v = lrelu(as_[(long long)s * 2 + head] + ad_[(long long)d * 2 + head]);
    float m = mono2f(mk[(long long)d * 2 + head]);
    float alpha = __expf(v - m) / ssum[(long long)d * 2 + head];

    const float4* hs = (const float4*)(H + (long long)s * 128);
    float4 hv = hs[lane];                          // coalesced 512B per wave
    float* op = out + (long long)d * 128 + lane * 4;
    unsafeAtomicAdd(op + 0, hv.x * alpha);
    unsafeAtomicAdd(op + 1, hv.y * alpha);
    unsafeAtomicAdd(op + 2, hv.z * alpha);
    unsafeAtomicAdd(op + 3, hv.w * alpha);
}

// ---------- 6) bias + ELU ----------
__global__ __launch_bounds__(256)
void bias_elu(float* __restrict__ Y, const float* __restrict__ B, long long total)
{
    long long i = (long long)blockIdx.x * blockDim.x + threadIdx.x;
    if (i >= total) return;
    float v = Y[i] + B[(int)(i & 127)];
    Y[i] = v > 0.f ? v : (__expf(v) - 1.f);
}

// ---------- host launcher ----------
extern "C" void kernel_launch(void* const* d_in, const int* in_sizes, int n_in,
                              void* d_out, int out_size, void* d_ws, size_t ws_size,
                              hipStream_t stream)
{
    const float* x     = (const float*)d_in[0];
    const int*   ei    = (const int*)  d_in[1];
    const float* W1    = (const float*)d_in[2];
    const float* asrc1 = (const float*)d_in[3];
    const float* adst1 = (const float*)d_in[4];
    const float* b1    = (const float*)d_in[5];
    const float* W2    = (const float*)d_in[6];
    const float* asrc2 = (const float*)d_in[7];
    const float* adst2 = (const float*)d_in[8];
    const float* b2    = (const float*)d_in[9];

    const int       N = in_sizes[0] / D1;          // IN_CH == D1 == 128
    const long long E = (long long)in_sizes[1] / 2;
    const long long Etot = E + N;
    const long long NF = (long long)N * D1;

    char* p = (char*)d_ws;
    float*    h    = (float*)p;    p += NF * sizeof(float);
    float*    xmid = (float*)p;    p += NF * sizeof(float);
    float*    as_  = (float*)p;    p += (size_t)N * 2 * sizeof(float);
    float*    ad_  = (float*)p;    p += (size_t)N * 2 * sizeof(float);
    unsigned* mk   = (unsigned*)p; p += (size_t)N * 2 * sizeof(unsigned);
    float*    ssum = (float*)p;    p += (size_t)N * 2 * sizeof(float);

    auto run_layer = [&](const float* xin, const float* W, const float* av,
                         const float* bv, const float* bias, float* outbuf) {
        hipMemsetAsync(mk,   0, (size_t)N * 2 * sizeof(unsigned), stream);
        hipMemsetAsync(ssum, 0, (size_t)N * 2 * sizeof(float),    stream);
        hipMemsetAsync(outbuf, 0, (size_t)NF * sizeof(float),     stream);

        gemm128_wmma<<<(unsigned)((N + 127) / 128), 256, 0, stream>>>(xin, W, h, N);
        node_alpha  <<<(unsigned)((N + 255) / 256), 256, 0, stream>>>(h, av, bv, as_, ad_, N);
        edge_max    <<<(unsigned)((Etot + 255) / 256), 256, 0, stream>>>(ei, as_, ad_, mk, E, Etot);
        edge_sum    <<<(unsigned)((Etot + 255) / 256), 256, 0, stream>>>(ei, as_, ad_, mk, ssum, E, Etot);
        edge_agg    <<<(unsigned)((Etot + 7) / 8),     256, 0, stream>>>(ei, as_, ad_, mk, ssum, h, outbuf, E, Etot);
        bias_elu    <<<(unsigned)((NF + 255) / 256),   256, 0, stream>>>(outbuf, bias, NF);
    };

    run_layer(x,    W1, asrc1, adst1, b1, xmid);
    run_layer(xmid, W2, asrc2, adst2, b2, (float*)d_out);
}